// GlobalPool_11287174053946
// MI455X (gfx1250) — compile-verified
//
#include <hip/hip_runtime.h>
#include <math.h>

#define NN 400000
#define BB 8192
#define FF 256
#define CH 1024

typedef float v2f __attribute__((ext_vector_type(2)));
typedef float v8f __attribute__((ext_vector_type(8)));

// -------------------- segment start offsets (lower bounds) --------------------
__global__ void __launch_bounds__(256) seg_starts_kernel(const int* __restrict__ seg,
                                                         int* __restrict__ start) {
  int i = blockIdx.x * blockDim.x + threadIdx.x;
  if (i >= NN) return;
  int s = seg[i];
  if (i == 0) {
    for (int b = 0; b <= s; ++b) start[b] = 0;
  } else {
    int p = seg[i - 1];
    for (int b = p + 1; b <= s; ++b) start[b] = i;
  }
  if (i == NN - 1) {
    for (int b = s + 1; b <= BB; ++b) start[b] = NN;
  }
}

// -------- fused per-graph: logits -> online softmax -> weighted feature sum ----
__global__ void __launch_bounds__(256) attn_pool_kernel(
    const float* __restrict__ x,   // node_feats N x F
    const float* __restrict__ g,   // g_feats    B x F
    const float* __restrict__ Wl,  // 1 x 2F
    const float* __restrict__ bl,  // 1
    const int* __restrict__ start, // B+1
    float* __restrict__ sbuf)      // B x F : sum_i softmax_i * x_i
{
  __shared__ float wlB[FF];
  __shared__ float zc[CH];
  __shared__ float red[256];
  __shared__ float sh_cg, sh_m, sh_d, sh_scale, sh_newm;

  const int b = blockIdx.x;
  const int t = threadIdx.x;
  const int lane = t & 31, wv = t >> 5;
  const int s0 = start[b], e0 = start[b + 1];
  const int len = e0 - s0;

  // graph-constant part of the logit: dot(relu(g_b), Wl[:F]) + bias
  wlB[t] = Wl[FF + t];
  float gv = g[(size_t)b * FF + t];
  red[t] = (gv > 0.f ? gv : 0.f) * Wl[t];
  __syncthreads();
  for (int off = 128; off > 0; off >>= 1) {
    if (t < off) red[t] += red[t + off];
    __syncthreads();
  }
  if (t == 0) { sh_cg = red[0] + bl[0]; sh_m = -INFINITY; sh_d = 0.f; }
  __syncthreads();
  const float cg = sh_cg;

  float facc = 0.f;  // accumulator for feature column t

  for (int i0 = 0; i0 < len; i0 += CH) {
    const int cn = min(CH, len - i0);

    // --- per-node logits for this chunk (one wave per node) ---
    for (int j = wv; j < cn; j += 8) {
      const float* xr = x + (size_t)(s0 + i0 + j) * FF;
      float a = 0.f;
#pragma unroll
      for (int k = 0; k < 8; ++k) a = fmaf(xr[lane + 32 * k], wlB[lane + 32 * k], a);
      a += __shfl_xor(a, 16, 32);
      a += __shfl_xor(a, 8, 32);
      a += __shfl_xor(a, 4, 32);
      a += __shfl_xor(a, 2, 32);
      a += __shfl_xor(a, 1, 32);
      if (lane == 0) {
        float zz = cg + a;
        zc[j] = zz > 0.f ? zz : 0.01f * zz;  // leaky_relu
      }
    }
    __syncthreads();

    // --- chunk max ---
    float lm = -INFINITY;
    for (int j = t; j < cn; j += 256) lm = fmaxf(lm, zc[j]);
    red[t] = lm;
    __syncthreads();
    for (int off = 128; off > 0; off >>= 1) {
      if (t < off) red[t] = fmaxf(red[t], red[t + off]);
      __syncthreads();
    }
    if (t == 0) {
      float nm = fmaxf(sh_m, red[0]);
      sh_scale = (sh_m == -INFINITY) ? 0.f : __expf(sh_m - nm);
      sh_newm = nm;
    }
    __syncthreads();
    const float sc = sh_scale, nm = sh_newm;
    facc *= sc;  // online-softmax rescale of running weighted sum

    // --- exponentials + chunk denom ---
    float ls = 0.f;
    for (int j = t; j < cn; j += 256) {
      float e = __expf(zc[j] - nm);
      zc[j] = e;
      ls += e;
    }
    red[t] = ls;
    __syncthreads();
    for (int off = 128; off > 0; off >>= 1) {
      if (t < off) red[t] += red[t + off];
      __syncthreads();
    }
    if (t == 0) { sh_d = sh_d * sc + red[0]; sh_m = nm; }
    __syncthreads();

    // --- weighted feature accumulation (column t, coalesced rows) ---
    const float* xp = x + (size_t)(s0 + i0) * FF + t;
    for (int j = 0; j < cn; ++j) {
      facc = fmaf(zc[j], xp[(size_t)j * FF], facc);
    }
    __syncthreads();
  }

  const float d = sh_d;
  sbuf[(size_t)b * FF + t] = (d > 0.f) ? facc / d : 0.f;
}

// ---------------- context = elu(s @ W_proj^T + b_proj), fp32 WMMA -------------
__global__ void __launch_bounds__(32) proj_elu_kernel(
    const float* __restrict__ S, const float* __restrict__ Wp,
    const float* __restrict__ bp, float* __restrict__ ctx) {
  const int m0 = blockIdx.x * 16;
  const int n0 = blockIdx.y * 16;
  const int lane = threadIdx.x;
  const int ln = lane & 15, hi = lane >> 4;

  v8f c = {};
  const float* arow = S + (size_t)(m0 + ln) * FF + 2 * hi;
  const float* brow = Wp + (size_t)(n0 + ln) * FF + 2 * hi;  // C[m,n]=sum_k S[m,k]*Wp[n,k]
  for (int k = 0; k < FF; k += 4) {
    v2f a = *(const v2f*)(arow + k);
    v2f bfr = *(const v2f*)(brow + k);
    c = __builtin_amdgcn_wmma_f32_16x16x4_f32(false, a, false, bfr, (short)0, c,
                                              false, false);
  }
#pragma unroll
  for (int r = 0; r < 8; ++r) {
    const int m = m0 + r + 8 * hi;
    const int n = n0 + ln;
    float v = c[r] + bp[n];
    ctx[(size_t)m * FF + n] = v > 0.f ? v : expm1f(v);  // elu(alpha=1)
  }
}

// ------------- fused GRU: 6 WMMA accumulators + gate epilogue -----------------
__device__ __forceinline__ float sigmoidf_(float v) { return 1.f / (1.f + __expf(-v)); }

__global__ void __launch_bounds__(32) gru_kernel(
    const float* __restrict__ ctx, const float* __restrict__ g,
    const float* __restrict__ Wih, const float* __restrict__ bih,
    const float* __restrict__ Whh, const float* __restrict__ bhh,
    float* __restrict__ out) {
  const int m0 = blockIdx.x * 16;
  const int n0 = blockIdx.y * 16;
  const int lane = threadIdx.x;
  const int ln = lane & 15, hi = lane >> 4;

  v8f a_ir = {}, a_iz = {}, a_in = {}, a_hr = {}, a_hz = {}, a_hn = {};

  const float* actx = ctx + (size_t)(m0 + ln) * FF + 2 * hi;
  const float* agf  = g   + (size_t)(m0 + ln) * FF + 2 * hi;
  const float* bir = Wih + (size_t)(0 * FF + n0 + ln) * FF + 2 * hi;
  const float* biz = Wih + (size_t)(1 * FF + n0 + ln) * FF + 2 * hi;
  const float* bin = Wih + (size_t)(2 * FF + n0 + ln) * FF + 2 * hi;
  const float* bhr = Whh + (size_t)(0 * FF + n0 + ln) * FF + 2 * hi;
  const float* bhz = Whh + (size_t)(1 * FF + n0 + ln) * FF + 2 * hi;
  const float* bhn = Whh + (size_t)(2 * FF + n0 + ln) * FF + 2 * hi;

  for (int k = 0; k < FF; k += 4) {
    v2f ac = *(const v2f*)(actx + k);
    v2f ag = *(const v2f*)(agf + k);
    a_ir = __builtin_amdgcn_wmma_f32_16x16x4_f32(false, ac, false, *(const v2f*)(bir + k),
                                                 (short)0, a_ir, false, false);
    a_iz = __builtin_amdgcn_wmma_f32_16x16x4_f32(false, ac, false, *(const v2f*)(biz + k),
                                                 (short)0, a_iz, false, false);
    a_in = __builtin_amdgcn_wmma_f32_16x16x4_f32(false, ac, false, *(const v2f*)(bin + k),
                                                 (short)0, a_in, false, false);
    a_hr = __builtin_amdgcn_wmma_f32_16x16x4_f32(false, ag, false, *(const v2f*)(bhr + k),
                                                 (short)0, a_hr, false, false);
    a_hz = __builtin_amdgcn_wmma_f32_16x16x4_f32(false, ag, false, *(const v2f*)(bhz + k),
                                                 (short)0, a_hz, false, false);
    a_hn = __builtin_amdgcn_wmma_f32_16x16x4_f32(false, ag, false, *(const v2f*)(bhn + k),
                                                 (short)0, a_hn, false, false);
  }

#pragma unroll
  for (int r = 0; r < 8; ++r) {
    const int m = m0 + r + 8 * hi;
    const int n = n0 + ln;
    float rv = sigmoidf_(a_ir[r] + bih[n] + a_hr[r] + bhh[n]);
    float uv = sigmoidf_(a_iz[r] + bih[FF + n] + a_hz[r] + bhh[FF + n]);
    float nv = tanhf(a_in[r] + bih[2 * FF + n] + rv * (a_hn[r] + bhh[2 * FF + n]));
    float hprev = g[(size_t)m * FF + n];
    out[(size_t)m * FF + n] = (1.f - uv) * nv + uv * hprev;
  }
}

// ------------------------------ launcher --------------------------------------
extern "C" void kernel_launch(void* const* d_in, const int* in_sizes, int n_in,
                              void* d_out, int out_size, void* d_ws, size_t ws_size,
                              hipStream_t stream) {
  const float* node_feats = (const float*)d_in[0];
  const float* g_feats    = (const float*)d_in[1];
  const int*   seg_ids    = (const int*)d_in[2];
  // d_in[3] = num_graphs (compile-time BB)
  const float* W_logits   = (const float*)d_in[4];
  const float* b_logits   = (const float*)d_in[5];
  const float* W_proj     = (const float*)d_in[6];
  const float* b_proj     = (const float*)d_in[7];
  const float* W_ih       = (const float*)d_in[8];
  const float* b_ih       = (const float*)d_in[9];
  const float* W_hh       = (const float*)d_in[10];
  const float* b_hh       = (const float*)d_in[11];
  float* out = (float*)d_out;

  // workspace layout
  int*   seg_start = (int*)d_ws;                                  // (B+1) ints
  float* s_buf     = (float*)((char*)d_ws + 64 * 1024);           // B*F
  float* ctx_buf   = s_buf + (size_t)BB * FF;                     // B*F

  seg_starts_kernel<<<(NN + 255) / 256, 256, 0, stream>>>(seg_ids, seg_start);

  attn_pool_kernel<<<BB, 256, 0, stream>>>(node_feats, g_feats, W_logits, b_logits,
                                           seg_start, s_buf);

  dim3 gg(BB / 16, FF / 16);
  proj_elu_kernel<<<gg, 32, 0, stream>>>(s_buf, W_proj, b_proj, ctx_buf);

  gru_kernel<<<gg, 32, 0, stream>>>(ctx_buf, g_feats, W_ih, b_ih, W_hh, b_hh, out);
}